// C2yModel_20340965114417
// MI455X (gfx1250) — compile-verified
//
#include <hip/hip_runtime.h>
#include <math.h>

typedef __attribute__((ext_vector_type(2))) float v2f;
typedef __attribute__((ext_vector_type(8))) float v8f;

#define NNODES 8192
#define NEDGES 262144
#define NREL   8
#define HID    512
#define OUTD   256
#define BOXD   1024
#define EMBD   256
#define KCAT   1280
#define NPG    128
#define NGRAPH 64

__device__ __forceinline__ void atomAddF(float* p, float v) {
  __hip_atomic_fetch_add(p, v, __ATOMIC_RELAXED, __HIP_MEMORY_SCOPE_AGENT);
}

// ---------------------------------------------------------------------------
// WMMA f32 16x16x4 GEMM:  C[M,NOUT] = A[M,K] * B[K,NOUT] (+bias | +=C)
// AMODE 0: dense A (ld=K).  AMODE 1: A = concat(box[ld=1024], emb[label][ld=256])
// One wave -> 16x64 tile (4 accumulators). 8 waves / block.
// B staged in LDS in 16 x NOUT chunks with K-PAIR INTERLEAVED layout:
//   Bs[((k>>1)*NOUT + col)*2 + (k&1)] = B[kc+k][col]
// so each WMMA B-fragment {B[k][col], B[k+1][col]} is ONE aligned ds_load_b64.
// ---------------------------------------------------------------------------
template <int NOUT, int AMODE, bool ACC>
__global__ __launch_bounds__(256) void gemm_wmma_k(
    const float* __restrict__ A, const float* __restrict__ emb,
    const int* __restrict__ label, const float* __restrict__ B,
    const float* __restrict__ bias, float* __restrict__ C, int M, int K) {
  __shared__ float Bs[16 * NOUT];
  const int lane = threadIdx.x & 31;
  const int wave = threadIdx.x >> 5;
  const int half = lane >> 4;   // 0: K={0,1}, 1: K={2,3} per WMMA f32 layout
  const int m    = lane & 15;
  constexpr int NT = NOUT / 64;
  const int waveId  = blockIdx.x * 8 + wave;
  const int tileM   = waveId / NT;
  const int colBase = (waveId % NT) * 64;
  const int row     = tileM * 16 + m;          // A row for this lane
  const int rowD    = tileM * 16 + 8 * half;   // C/D row base for this half

  int lrow = 0;
  if (AMODE == 1) lrow = label[row];

  v8f acc[4];
#pragma unroll
  for (int t = 0; t < 4; ++t) {
    const int col = colBase + t * 16 + m;
    if (ACC) {
#pragma unroll
      for (int v = 0; v < 8; ++v)
        acc[t][v] = C[(size_t)(rowD + v) * NOUT + col];
    } else {
      const float bv = bias[col];
#pragma unroll
      for (int v = 0; v < 8; ++v) acc[t][v] = bv;
    }
  }

  // staging decomposition: quads of (row-pair pr, 4 columns cq)
  constexpr int QUADS   = (16 / 2) * (NOUT / 4);   // per chunk
  constexpr int QPERTHR = QUADS / 256;
  const int q0 = threadIdx.x * QPERTHR;

  for (int kc = 0; kc < K; kc += 16) {
    __syncthreads();
#pragma unroll
    for (int p = 0; p < QPERTHR; ++p) {
      const int q  = q0 + p;
      const int pr = q / (NOUT / 4);        // row pair 0..7
      const int cq = (q % (NOUT / 4)) * 4;  // column quad
      const float4 f0 = *(const float4*)(&B[(size_t)(kc + 2 * pr + 0) * NOUT + cq]);
      const float4 f1 = *(const float4*)(&B[(size_t)(kc + 2 * pr + 1) * NOUT + cq]);
      float* bp = &Bs[(pr * NOUT + cq) * 2];
      ((v2f*)bp)[0] = v2f{f0.x, f1.x};
      ((v2f*)bp)[1] = v2f{f0.y, f1.y};
      ((v2f*)bp)[2] = v2f{f0.z, f1.z};
      ((v2f*)bp)[3] = v2f{f0.w, f1.w};
      if (kc + 16 < K) {
        __builtin_prefetch(&B[(size_t)(kc + 16 + 2 * pr) * NOUT + cq], 0, 0);
        __builtin_prefetch(&B[(size_t)(kc + 17 + 2 * pr) * NOUT + cq], 0, 0);
      }
    }
    __syncthreads();
#pragma unroll
    for (int kk = 0; kk < 16; kk += 4) {
      const int kb = kc + kk + 2 * half;
      v2f a;
      if (AMODE == 0) {
        a = *(const v2f*)(&A[(size_t)row * K + kb]);
      } else {
        if (kb < BOXD) a = *(const v2f*)(&A[(size_t)row * BOXD + kb]);
        else           a = *(const v2f*)(&emb[(size_t)lrow * EMBD + (kb - BOXD)]);
      }
      const int prBase = (kk >> 1) + half;    // interleaved row-pair index
#pragma unroll
      for (int t = 0; t < 4; ++t) {
        const int col = colBase + t * 16 + m;
        const v2f b = *(const v2f*)(&Bs[(prBase * NOUT + col) * 2]);
        acc[t] = __builtin_amdgcn_wmma_f32_16x16x4_f32(
            false, a, false, b, (short)0, acc[t], false, false);
      }
    }
  }

#pragma unroll
  for (int t = 0; t < 4; ++t) {
    const int col = colBase + t * 16 + m;
#pragma unroll
    for (int v = 0; v < 8; ++v)
      C[(size_t)(rowD + v) * NOUT + col] = acc[t][v];
  }
}

// per-(relation,dst) edge counts (as float, exact < 2^24)
__global__ void count_edges_k(const int* __restrict__ et,
                              const int* __restrict__ dst,
                              float* __restrict__ cnt) {
  const int e = blockIdx.x * blockDim.x + threadIdx.x;
  if (e < NEDGES) atomAddF(&cnt[et[e] * NNODES + dst[e]], 1.0f);
}

// input-domain scatter: S[dst] += X[src] for edges of relation `rel`
__global__ __launch_bounds__(256) void scatter_k(
    const float* __restrict__ X, const int* __restrict__ src,
    const int* __restrict__ dst, const int* __restrict__ et, int rel,
    float* __restrict__ S) {
  const int e = blockIdx.x;
  if (et[e] != rel) return;
  const float* xs = X + (size_t)src[e] * HID;
  float*       sd = S + (size_t)dst[e] * HID;
  for (int j = threadIdx.x; j < HID; j += 256) atomAddF(&sd[j], xs[j]);
}

__global__ void normalize_k(float* __restrict__ S,
                            const float* __restrict__ cntR) {
  const int i = blockIdx.x * blockDim.x + threadIdx.x;  // NNODES*HID threads
  const float c = cntR[i >> 9];                          // HID == 512
  S[i] *= (1.0f / fmaxf(c, 1.0f));
}

__global__ void relu_k(float* __restrict__ x, int n) {
  const int i = blockIdx.x * blockDim.x + threadIdx.x;
  if (i < n) x[i] = fmaxf(x[i], 0.0f);
}

__global__ __launch_bounds__(256) void pool_k(const float* __restrict__ h,
                                              float* __restrict__ pooled) {
  const int b = blockIdx.x, j = threadIdx.x;  // 64 blocks x 256 threads
  float s = 0.0f;
  const float* base = h + (size_t)b * NPG * OUTD + j;
  for (int i = 0; i < NPG; ++i) s += base[(size_t)i * OUTD];
  pooled[b * OUTD + j] = s * (1.0f / (float)NPG);
}

__global__ __launch_bounds__(256) void fc1_k(const float* __restrict__ pooled,
                                             const float* __restrict__ Wc1,
                                             const float* __restrict__ bc1,
                                             float* __restrict__ t) {
  __shared__ float pr[OUTD];
  const int b = blockIdx.x, j = threadIdx.x;
  pr[j] = pooled[b * OUTD + j];
  __syncthreads();
  float s = bc1[j];
  for (int k = 0; k < OUTD; ++k) s += pr[k] * Wc1[k * OUTD + j];
  t[b * OUTD + j] = s;
}

__global__ void head_k(const float* __restrict__ t,
                       const float* __restrict__ Wc2,
                       const float* __restrict__ bc2,
                       float* __restrict__ out) {
  const int b = threadIdx.x;
  if (b >= NGRAPH) return;
  float l0 = bc2[0], l1 = bc2[1];
  const float* tr = t + (size_t)b * OUTD;
  for (int k = 0; k < OUTD; ++k) {
    l0 += tr[k] * Wc2[k * 2 + 0];
    l1 += tr[k] * Wc2[k * 2 + 1];
  }
  out[b * 2 + 0] = l0;
  out[b * 2 + 1] = l1;
  const float mx = fmaxf(l0, l1);
  const float e0 = __expf(l0 - mx), e1 = __expf(l1 - mx);
  const float inv = 1.0f / (e0 + e1);
  out[2 * NGRAPH + b * 2 + 0] = e0 * inv;
  out[2 * NGRAPH + b * 2 + 1] = e1 * inv;
}

extern "C" void kernel_launch(void* const* d_in, const int* in_sizes, int n_in,
                              void* d_out, int out_size, void* d_ws,
                              size_t ws_size, hipStream_t stream) {
  (void)in_sizes; (void)n_in; (void)out_size; (void)ws_size;
  const float* box    = (const float*)d_in[0];
  const int*   label  = (const int*)d_in[1];
  const int*   src    = (const int*)d_in[2];
  const int*   dst    = src + NEDGES;
  const int*   etype  = (const int*)d_in[3];
  const float* emb    = (const float*)d_in[5];
  const float* Wlin   = (const float*)d_in[6];
  const float* blin   = (const float*)d_in[7];
  const float* relW1  = (const float*)d_in[8];
  const float* rootW1 = (const float*)d_in[9];
  const float* b1     = (const float*)d_in[10];
  const float* relW2  = (const float*)d_in[11];
  const float* rootW2 = (const float*)d_in[12];
  const float* b2     = (const float*)d_in[13];
  const float* Wc1    = (const float*)d_in[14];
  const float* bc1    = (const float*)d_in[15];
  const float* Wc2    = (const float*)d_in[16];
  const float* bc2    = (const float*)d_in[17];
  float* out = (float*)d_out;

  float* ws  = (float*)d_ws;
  float* h1  = ws;                             // 8192*512
  float* h2  = h1 + (size_t)NNODES * HID;      // 8192*512
  float* h3  = h2 + (size_t)NNODES * HID;      // 8192*256
  float* S   = h3 + (size_t)NNODES * OUTD;     // 8192*512
  float* cnt = S + (size_t)NNODES * HID;       // 8*8192
  float* tcl = cnt + (size_t)NREL * NNODES;    // 64*256
  float* pooled = tcl + NGRAPH * OUTD;         // 64*256

  hipMemsetAsync(cnt, 0, (size_t)NREL * NNODES * sizeof(float), stream);
  count_edges_k<<<NEDGES / 256, 256, 0, stream>>>(etype, dst, cnt);

  // h1 = concat(box, emb[label]) @ Wlin + blin
  gemm_wmma_k<HID, 1, false><<<(NNODES / 16) * (HID / 64) / 8, 256, 0, stream>>>(
      box, emb, label, Wlin, blin, h1, NNODES, KCAT);

  // conv1: h2 = relu( h1@rootW1 + b1 + sum_r mean_r(h1)@W1_r )
  gemm_wmma_k<HID, 0, false><<<(NNODES / 16) * (HID / 64) / 8, 256, 0, stream>>>(
      h1, nullptr, nullptr, rootW1, b1, h2, NNODES, HID);
  for (int r = 0; r < NREL; ++r) {
    hipMemsetAsync(S, 0, (size_t)NNODES * HID * sizeof(float), stream);
    scatter_k<<<NEDGES, 256, 0, stream>>>(h1, src, dst, etype, r, S);
    normalize_k<<<NNODES * HID / 256, 256, 0, stream>>>(S, cnt + (size_t)r * NNODES);
    gemm_wmma_k<HID, 0, true><<<(NNODES / 16) * (HID / 64) / 8, 256, 0, stream>>>(
        S, nullptr, nullptr, relW1 + (size_t)r * HID * HID, nullptr, h2, NNODES, HID);
  }
  relu_k<<<NNODES * HID / 256, 256, 0, stream>>>(h2, NNODES * HID);

  // conv2: h3 = h2@rootW2 + b2 + sum_r mean_r(h2)@W2_r
  gemm_wmma_k<OUTD, 0, false><<<(NNODES / 16) * (OUTD / 64) / 8, 256, 0, stream>>>(
      h2, nullptr, nullptr, rootW2, b2, h3, NNODES, HID);
  for (int r = 0; r < NREL; ++r) {
    hipMemsetAsync(S, 0, (size_t)NNODES * HID * sizeof(float), stream);
    scatter_k<<<NEDGES, 256, 0, stream>>>(h2, src, dst, etype, r, S);
    normalize_k<<<NNODES * HID / 256, 256, 0, stream>>>(S, cnt + (size_t)r * NNODES);
    gemm_wmma_k<OUTD, 0, true><<<(NNODES / 16) * (OUTD / 64) / 8, 256, 0, stream>>>(
        S, nullptr, nullptr, relW2 + (size_t)r * HID * OUTD, nullptr, h3, NNODES, HID);
  }

  pool_k<<<NGRAPH, OUTD, 0, stream>>>(h3, pooled);
  fc1_k<<<NGRAPH, OUTD, 0, stream>>>(pooled, Wc1, bc1, tcl);
  head_k<<<1, 64, 0, stream>>>(tcl, Wc2, bc2, out);
}